// SelectiveKernelAttn_11776800325704
// MI455X (gfx1250) — compile-verified
//
#include <hip/hip_runtime.h>
#include <hip/hip_bf16.h>

typedef __attribute__((ext_vector_type(2))) float v2f;
typedef __attribute__((ext_vector_type(8))) float v8f;

#define BN_EPS 1e-5f
#define B_  64
#define P_  2
#define C_  512
#define HW_ 1024
#define A_  32
#define K_  (C_ * P_)   // 1024

// ---------------------------------------------------------------------------
// Kernel 1 (bandwidth bound, ~256 MB read): s[b,c] = (sum over p,h,w) / 1024
// One wave32 per (b,c); lane does float4 loads, 8 per path, fully coalesced.
// ---------------------------------------------------------------------------
__global__ void sk_reduce_kernel(const float* __restrict__ x,
                                 float* __restrict__ s) {
  const int wave = (blockIdx.x * blockDim.x + threadIdx.x) >> 5;  // 0..32767
  const int lane = threadIdx.x & 31;
  const int b = wave >> 9;     // /512
  const int c = wave & 511;

  float acc = 0.f;
#pragma unroll
  for (int p = 0; p < P_; ++p) {
    const float4* base = reinterpret_cast<const float4*>(
        x + (size_t)((b * P_ + p) * C_ + c) * HW_);
#pragma unroll
    for (int i = 0; i < 8; ++i) {
      float4 v = base[lane + i * 32];
      acc += (v.x + v.y) + (v.z + v.w);
    }
  }
  // wave32 butterfly reduction
#pragma unroll
  for (int off = 16; off > 0; off >>= 1) acc += __shfl_xor(acc, off, 32);
  if (lane == 0) s[b * C_ + c] = acc * (1.0f / 1024.0f);
}

// ---------------------------------------------------------------------------
// Kernel 2a: z = s[64,512] @ w_reduce[32,512]^T  (M=64,N=32,K=512) via
// v_wmma_f32_16x16x4_f32, fused BN(inference) + ReLU -> zr[64,32].
// 8 waves, one 16x16 tile each; 128 K-steps of 4.
// ---------------------------------------------------------------------------
__global__ void sk_gemm1_bn_kernel(const float* __restrict__ s,
                                   const float* __restrict__ w_reduce,
                                   const float* __restrict__ gamma,
                                   const float* __restrict__ beta,
                                   const float* __restrict__ mean,
                                   const float* __restrict__ var,
                                   float* __restrict__ zr) {
  const int wave = threadIdx.x >> 5;   // 0..7
  const int lane = threadIdx.x & 31;
  const int m    = lane & 15;
  const int hi   = lane >> 4;
  const int koff = hi * 2;
  const int M0 = (wave >> 1) * 16;     // 0,16,32,48
  const int N0 = (wave & 1) * 16;      // 0,16

  v8f acc = {};
  for (int k = 0; k < C_; k += 4) {
    // A frag: rows of s; lanes 0-15 hold K=k,k+1, lanes 16-31 hold K=k+2,k+3
    v2f a = *(const v2f*)(s + (M0 + m) * C_ + k + koff);
    // B frag: B = w_reduce^T, so B[kk][n] = w_reduce[n][kk] (contig float2)
    v2f b = *(const v2f*)(w_reduce + (N0 + m) * C_ + k + koff);
    acc = __builtin_amdgcn_wmma_f32_16x16x4_f32(false, a, false, b,
                                                (short)0, acc, false, false);
  }

  const int   n  = N0 + m;                       // BN channel = output column
  const float sc = gamma[n] * rsqrtf(var[n] + BN_EPS);
  const float mu = mean[n];
  const float bt = beta[n];
#pragma unroll
  for (int r = 0; r < 8; ++r) {
    float z = (acc[r] - mu) * sc + bt;           // row = M0 + r + 8*hi
    z = fmaxf(z, 0.f);                           // ReLU
    zr[(M0 + r + 8 * hi) * A_ + n] = z;
  }
}

// ---------------------------------------------------------------------------
// Kernel 2b: a = zr[64,32] @ w_select[1024,32]^T (M=64,N=1024,K=32).
// 256 waves, one 16x16 tile each; 8 K-steps of 4.
// ---------------------------------------------------------------------------
__global__ void sk_gemm2_kernel(const float* __restrict__ zr,
                                const float* __restrict__ w_select,
                                float* __restrict__ aout) {
  const int wave = (blockIdx.x * blockDim.x + threadIdx.x) >> 5;  // 0..255
  const int lane = threadIdx.x & 31;
  const int m    = lane & 15;
  const int hi   = lane >> 4;
  const int koff = hi * 2;
  const int M0 = (wave & 3) * 16;      // 4 M-tiles
  const int N0 = (wave >> 2) * 16;     // 64 N-tiles

  v8f acc = {};
#pragma unroll
  for (int k = 0; k < A_; k += 4) {
    v2f a = *(const v2f*)(zr + (M0 + m) * A_ + k + koff);
    v2f b = *(const v2f*)(w_select + (N0 + m) * A_ + k + koff);
    acc = __builtin_amdgcn_wmma_f32_16x16x4_f32(false, a, false, b,
                                                (short)0, acc, false, false);
  }
#pragma unroll
  for (int r = 0; r < 8; ++r)
    aout[(M0 + r + 8 * hi) * K_ + N0 + m] = acc[r];
}

// ---------------------------------------------------------------------------
// Kernel 3: softmax over path dim (P=2): pairs (a[b,c], a[b,512+c]).
// ---------------------------------------------------------------------------
__global__ void sk_softmax_kernel(const float* __restrict__ aout,
                                  float* __restrict__ out) {
  const int t = blockIdx.x * blockDim.x + threadIdx.x;  // 0..32767
  const int b = t >> 9;
  const int c = t & 511;
  float a0 = aout[b * K_ + c];
  float a1 = aout[b * K_ + C_ + c];
  float mx = fmaxf(a0, a1);
  float e0 = __expf(a0 - mx);
  float e1 = __expf(a1 - mx);
  float inv = 1.f / (e0 + e1);
  out[b * K_ + c]       = e0 * inv;   // p = 0
  out[b * K_ + C_ + c]  = e1 * inv;   // p = 1
}

// ---------------------------------------------------------------------------
extern "C" void kernel_launch(void* const* d_in, const int* in_sizes, int n_in,
                              void* d_out, int out_size, void* d_ws,
                              size_t ws_size, hipStream_t stream) {
  const float* x        = (const float*)d_in[0];
  const float* w_reduce = (const float*)d_in[1];
  const float* gamma    = (const float*)d_in[2];
  const float* beta     = (const float*)d_in[3];
  const float* mean     = (const float*)d_in[4];
  const float* var      = (const float*)d_in[5];
  const float* w_select = (const float*)d_in[6];
  float* out = (float*)d_out;

  // workspace layout (floats): s[64*512] | zr[64*32] | a[64*1024]  (~400 KB)
  float* s    = (float*)d_ws;
  float* zr   = s + B_ * C_;
  float* aout = zr + B_ * A_;

  // 32768 waves / 8 per block = 4096 blocks
  sk_reduce_kernel<<<4096, 256, 0, stream>>>(x, s);
  sk_gemm1_bn_kernel<<<1, 256, 0, stream>>>(s, w_reduce, gamma, beta, mean,
                                            var, zr);
  sk_gemm2_kernel<<<32, 256, 0, stream>>>(zr, w_select, aout);
  sk_softmax_kernel<<<128, 256, 0, stream>>>(aout, out);
}